// Regionloss_67362267070733
// MI455X (gfx1250) — compile-verified
//
#include <hip/hip_runtime.h>

// Region loss: weighted L1 with per-batch 40th-percentile threshold on channel mean.
// Shapes: y_true, y_pred = (16, 3, 512, 512) fp32.  Output: 1 fp32 scalar (mean).
//
// Pipeline (all on `stream`, graph-capture safe):
//   k_init  : zero histograms/state/out
//   3x (k_hist -> k_scan) : MSB-first radix select (11/11/10 bits) of the
//             k-th smallest gray value per batch (k = 104856), exact fp32 thr.
//   k_final : fused weighted-L1 reduction; wave reduce via V_WMMA_F32_16X16X4_F32.

#define BATCH 16
#define HW    (512 * 512)       // 262144
#define HW4   (HW / 4)          // 65536 float4-groups per (batch,channel) plane
#define NB    2048              // histogram bins (max digit width 11 bits)
#define KIDX  104856u           // int(512*512*0.4 - 1)
#define NTOT  12582912.0f       // 16*3*512*512

typedef float v2f __attribute__((ext_vector_type(2)));
typedef float v8f __attribute__((ext_vector_type(8)));

// Order-preserving float <-> uint key mapping
__device__ __forceinline__ unsigned f2key(float f) {
    unsigned u = __float_as_uint(f);
    return (u & 0x80000000u) ? ~u : (u | 0x80000000u);
}
__device__ __forceinline__ float key2f(unsigned k) {
    unsigned u = (k & 0x80000000u) ? (k & 0x7FFFFFFFu) : ~k;
    return __uint_as_float(u);
}

// Workspace layout (u32 words):
//   [0, BATCH*NB)          per-batch histogram
//   [H, H+16)              prefix (key bits resolved so far)
//   [H+16, H+32)           remaining rank
//   [H+32, H+48)           thr (float bits)
#define WS_HIST   0
#define WS_PREFIX (BATCH * NB)
#define WS_RANK   (BATCH * NB + 16)
#define WS_THR    (BATCH * NB + 32)
#define WS_WORDS  (BATCH * NB + 48)

__global__ void k_init(unsigned* __restrict__ ws, float* __restrict__ out) {
    int stride = gridDim.x * blockDim.x;
    for (int i = blockIdx.x * blockDim.x + threadIdx.x; i < WS_WORDS; i += stride) {
        unsigned v = 0u;
        if (i >= WS_RANK && i < WS_RANK + BATCH) v = KIDX;
        ws[i] = v;
    }
    if (blockIdx.x == 0 && threadIdx.x == 0) out[0] = 0.0f;
}

__global__ void k_hist(const float* __restrict__ pred, unsigned* __restrict__ ws,
                       int shift, unsigned dmask, unsigned himask) {
    unsigned* __restrict__ hist = ws + WS_HIST;
    const unsigned* __restrict__ prefix = ws + WS_PREFIX;
    const int G = BATCH * HW4;           // 1,048,576 float4 groups
    const int stride = gridDim.x * blockDim.x;
    for (int g = blockIdx.x * blockDim.x + threadIdx.x; g < G; g += stride) {
        int b  = g >> 16;                // HW4 == 2^16
        int gi = g & (HW4 - 1);
        const float4* p = reinterpret_cast<const float4*>(pred) + (size_t)b * 3 * HW4 + gi;
        float4 c0 = p[0];
        float4 c1 = p[HW4];
        float4 c2 = p[2 * HW4];
        unsigned pref = prefix[b];
        unsigned* hb = hist + (b << 11);
        unsigned k0 = f2key((c0.x + c1.x + c2.x) / 3.0f);
        unsigned k1 = f2key((c0.y + c1.y + c2.y) / 3.0f);
        unsigned k2 = f2key((c0.z + c1.z + c2.z) / 3.0f);
        unsigned k3 = f2key((c0.w + c1.w + c2.w) / 3.0f);
        if (((k0 ^ pref) & himask) == 0u) atomicAdd(&hb[(k0 >> shift) & dmask], 1u);
        if (((k1 ^ pref) & himask) == 0u) atomicAdd(&hb[(k1 >> shift) & dmask], 1u);
        if (((k2 ^ pref) & himask) == 0u) atomicAdd(&hb[(k2 >> shift) & dmask], 1u);
        if (((k3 ^ pref) & himask) == 0u) atomicAdd(&hb[(k3 >> shift) & dmask], 1u);
    }
}

// One block per batch. Thread 0 serially scans (<=2048 bins; negligible),
// then everyone re-zeroes this batch's histogram for the next pass.
__global__ void k_scan(unsigned* __restrict__ ws, int shift, int nbins, int last) {
    int b = blockIdx.x;
    unsigned* hist   = ws + WS_HIST + (b << 11);
    unsigned* prefix = ws + WS_PREFIX;
    unsigned* rank   = ws + WS_RANK;
    unsigned* thr    = ws + WS_THR;
    if (threadIdx.x == 0) {
        unsigned r = rank[b], cum = 0u;
        int sel = nbins - 1;
        for (int d = 0; d < nbins; ++d) {
            unsigned c = hist[d];
            if (cum + c > r) { sel = d; break; }
            cum += c;
        }
        unsigned p = prefix[b] | ((unsigned)sel << shift);
        prefix[b] = p;
        rank[b]   = r - cum;
        if (last) thr[b] = __float_as_uint(key2f(p));
    }
    __syncthreads();
    for (int d = threadIdx.x; d < NB; d += blockDim.x) hist[d] = 0u;
}

__global__ void k_final(const float* __restrict__ ytrue, const float* __restrict__ ypred,
                        const unsigned* __restrict__ ws, float* __restrict__ out) {
    const unsigned* __restrict__ thr = ws + WS_THR;
    float acc = 0.0f;
    const int G = BATCH * HW4;
    const int stride = gridDim.x * blockDim.x;   // grid sized so trip count is uniform
    for (int g = blockIdx.x * blockDim.x + threadIdx.x; g < G; g += stride) {
        int b  = g >> 16;
        int gi = g & (HW4 - 1);
        size_t base = (size_t)b * 3 * HW4 + gi;
        const float4* p = reinterpret_cast<const float4*>(ypred) + base;
        const float4* t = reinterpret_cast<const float4*>(ytrue) + base;
        float4 p0 = p[0], p1 = p[HW4], p2 = p[2 * HW4];
        float4 t0 = t[0], t1 = t[HW4], t2 = t[2 * HW4];
        float th = __uint_as_float(thr[b]);

        float gr, w;
        gr = (p0.x + p1.x + p2.x) / 3.0f; w = (gr <= th) ? 0.8f : 0.2f;
        acc += w * (fabsf(p0.x - t0.x) + fabsf(p1.x - t1.x) + fabsf(p2.x - t2.x));
        gr = (p0.y + p1.y + p2.y) / 3.0f; w = (gr <= th) ? 0.8f : 0.2f;
        acc += w * (fabsf(p0.y - t0.y) + fabsf(p1.y - t1.y) + fabsf(p2.y - t2.y));
        gr = (p0.z + p1.z + p2.z) / 3.0f; w = (gr <= th) ? 0.8f : 0.2f;
        acc += w * (fabsf(p0.z - t0.z) + fabsf(p1.z - t1.z) + fabsf(p2.z - t2.z));
        gr = (p0.w + p1.w + p2.w) / 3.0f; w = (gr <= th) ? 0.8f : 0.2f;
        acc += w * (fabsf(p0.w - t0.w) + fabsf(p1.w - t1.w) + fabsf(p2.w - t2.w));
    }

    // Wave-level reduction via V_WMMA_F32_16X16X4_F32:
    //   A(16x4): lane L<16 -> A[L][0]=acc, A[L][1]=0;  lane L>=16 -> A[L-16][2]=acc, [3]=0
    //   B(4x16) = ones  =>  D[m][n] = acc[m] + acc[m+16]
    //   Lanes 0-15 hold rows 0-7 across D[0..7]; lanes 16-31 hold rows 8-15.
    v2f A;  A[0] = acc;  A[1] = 0.0f;
    v2f Bv; Bv[0] = 1.0f; Bv[1] = 1.0f;
    v8f C = {};
    v8f D = __builtin_amdgcn_wmma_f32_16x16x4_f32(
        /*neg_a=*/false, A, /*neg_b=*/false, Bv,
        /*c_mod=*/(short)0, C, /*reuse_a=*/false, /*reuse_b=*/false);
    float h = D[0] + D[1] + D[2] + D[3] + D[4] + D[5] + D[6] + D[7];
    h += __shfl_xor(h, 16, 32);          // both halves now hold the full 32-lane sum

    __shared__ float warpSums[8];
    int lane = threadIdx.x & 31, wid = threadIdx.x >> 5;
    if (lane == 0) warpSums[wid] = h;
    __syncthreads();
    if (threadIdx.x == 0) {
        float s = 0.0f;
        int nw = blockDim.x >> 5;
        for (int i = 0; i < nw; ++i) s += warpSums[i];
        atomicAdd(out, s * (1.0f / NTOT));
    }
}

extern "C" void kernel_launch(void* const* d_in, const int* in_sizes, int n_in,
                              void* d_out, int out_size, void* d_ws, size_t ws_size,
                              hipStream_t stream) {
    const float* y_true = (const float*)d_in[0];
    const float* y_pred = (const float*)d_in[1];
    float* out  = (float*)d_out;
    unsigned* ws = (unsigned*)d_ws;   // needs WS_WORDS*4 ≈ 132 KB of scratch

    k_init<<<256, 256, 0, stream>>>(ws, out);

    // Radix-select passes: digit widths 11, 11, 10 (MSB first).
    k_hist<<<1024, 256, 0, stream>>>(y_pred, ws, 21, 0x7FFu, 0x00000000u);
    k_scan<<<BATCH, 256, 0, stream>>>(ws, 21, 2048, 0);

    k_hist<<<1024, 256, 0, stream>>>(y_pred, ws, 10, 0x7FFu, 0xFFE00000u);
    k_scan<<<BATCH, 256, 0, stream>>>(ws, 10, 2048, 0);

    k_hist<<<1024, 256, 0, stream>>>(y_pred, ws, 0, 0x3FFu, 0xFFFFFC00u);
    k_scan<<<BATCH, 256, 0, stream>>>(ws, 0, 1024, 1);

    // Fused weighted-L1 mean: 1024*256 threads * 4 groups = 1,048,576 groups exactly
    // (uniform trip count -> EXEC all-ones at the WMMA reduction).
    k_final<<<1024, 256, 0, stream>>>(y_true, y_pred, ws, out);
}